// yolo_19636590477745
// MI455X (gfx1250) — compile-verified
//
#include <hip/hip_runtime.h>

// YOLO head decode (pure elementwise, bandwidth-bound).
// Input  : prediction (16, 255, 76, 76) fp32, viewed as (16,3,76,76,85)
// Output : predx|predy|predw|predh|conf (each 16*3*76*76) ++ classes (…*80),
//          concatenated flat in that order.
//
// Per cell vector of 85 consecutive floats:
//   e==0 : sigmoid(x) + h        e==1 : sigmoid(x) + w
//   e==2 : exp(x) * anchor_w/8   e==3 : exp(x) * anchor_h/8
//   e==4 : sigmoid(x)            e>=5 : sigmoid(x)  (class logits)

typedef float f32x4 __attribute__((ext_vector_type(4)));  // native vector type

namespace {

constexpr unsigned kNV     = 16u * 3u * 76u * 76u;  // 277,248 grid cells
constexpr unsigned kClsOff = 5u * kNV;              // start of class block

__device__ __forceinline__ float fsig(float x) {
  // 1/(1+exp(-x)) via v_exp_f32 + v_rcp_f32 (~2 ulp, TRANS unit)
  return __builtin_amdgcn_rcpf(1.0f + __expf(-x));
}

}  // namespace

__global__ __launch_bounds__(256) void yolo_decode_kernel(
    const float* __restrict__ in, const float* __restrict__ anch,
    float* __restrict__ out, unsigned n4) {
  unsigned tid = blockIdx.x * 256u + threadIdx.x;
  if (tid >= n4) return;

  // 16B-aligned b128 load of 4 consecutive prediction floats
  const f32x4 x4 = reinterpret_cast<const f32x4*>(in)[tid];
  unsigned base = tid * 4u;
  unsigned v0   = base / 85u;         // cell index  (magic-mul div)
  unsigned e0   = base - v0 * 85u;    // element within cell

  // Fast path: all 4 elements are class logits of the same cell and the
  // destination (cls + v0*80 + e0-5) is 16B aligned -> one b128 NT store.
  if (e0 >= 5u && e0 <= 81u && ((e0 - 5u) & 3u) == 0u) {
    f32x4 r;
    r.x = fsig(x4.x);
    r.y = fsig(x4.y);
    r.z = fsig(x4.z);
    r.w = fsig(x4.w);
    __builtin_nontemporal_store(
        r, reinterpret_cast<f32x4*>(out + kClsOff + v0 * 80u + (e0 - 5u)));
    return;
  }

  // General path: up to one cell-boundary crossing inside the quad.
  const float xs[4] = {x4.x, x4.y, x4.z, x4.w};
#pragma unroll
  for (int j = 0; j < 4; ++j) {
    unsigned e = e0 + (unsigned)j;
    unsigned v = v0;
    if (e >= 85u) { e -= 85u; ++v; }
    float x = xs[j];
    if (e >= 5u) {                               // class logits
      __builtin_nontemporal_store(fsig(x), out + kClsOff + v * 80u + (e - 5u));
    } else if (e == 0u) {                        // predx = sigmoid + h
      float h = (float)(v % 76u);
      __builtin_nontemporal_store(fsig(x) + h, out + v);
    } else if (e == 1u) {                        // predy = sigmoid + w
      float w = (float)((v / 76u) % 76u);
      __builtin_nontemporal_store(fsig(x) + w, out + kNV + v);
    } else if (e == 4u) {                        // conf
      __builtin_nontemporal_store(fsig(x), out + 4u * kNV + v);
    } else {                                     // e==2 (w) / e==3 (h)
      unsigned a = (v / 5776u) % 3u;             // anchor slot (MASK = 0,1,2)
      float anc  = anch[2u * a + (e - 2u)] * 0.125f;  // anchors / stride(8)
      __builtin_nontemporal_store(__expf(x) * anc, out + e * kNV + v);
    }
  }
}

extern "C" void kernel_launch(void* const* d_in, const int* in_sizes, int n_in,
                              void* d_out, int out_size, void* d_ws,
                              size_t ws_size, hipStream_t stream) {
  (void)n_in; (void)out_size; (void)d_ws; (void)ws_size;
  const float* in   = (const float*)d_in[0];   // prediction
  const float* anch = (const float*)d_in[2];   // anchors (d_in[1] = empty gt)
  float* out = (float*)d_out;

  unsigned n  = (unsigned)in_sizes[0];         // 16*255*76*76 = 23,566,080
  unsigned n4 = n / 4u;                        // exactly divisible by 4
  unsigned blocks = (n4 + 255u) / 256u;        // 256 thr = 8 wave32 / block
  yolo_decode_kernel<<<blocks, 256, 0, stream>>>(in, anch, out, n4);
}